// CubePad_71528385347689
// MI455X (gfx1250) — compile-verified
//
#include <hip/hip_runtime.h>

// ---------------------------------------------------------------------------
// CubePad for MI455X (gfx1250).
// Pure data movement: ~405 MB total traffic, HBM-bound (floor ~17us @23.3TB/s).
// Interior copy uses the CDNA5 async path:
//   global_load_async_to_lds_b128 (ASYNCcnt, th:NT)
//   -> stepwise s_wait_asynccnt N (async loads complete in order, so
//      cnt<=7-k proves slot k landed) -> ds_load_b128 -> global_store_b128.
// Async loads write LDS without VGPR return slots: 8x128B in flight per lane
// at minimal register pressure, and the drain overlaps in-flight loads.
// ---------------------------------------------------------------------------

#define C      64
#define H      256
#define W      256
#define HP     258
#define WP     258
#define IN_PLANE   (H * W)      // 65536 floats
#define OUT_PLANE  (HP * WP)    // 66564 floats
#define NPLANES    768          // 12 * 64
#define RING       1028         // 2*258 + 2*256 halo elements per plane

typedef float f4 __attribute__((ext_vector_type(4)));

// ---------------------------------------------------------------------------
// Interior: out[p, y+1, x+1] = in[p, y, x].
// Each block: one chunk of 2048 float4-quads = 32 rows of one plane.
// quads/plane = 256*64 = 16384 -> 8 chunks/plane, 768*8 = 6144 blocks.
// ---------------------------------------------------------------------------
__global__ __launch_bounds__(256) void cubepad_interior(
    const float* __restrict__ in, float* __restrict__ out) {
  __shared__ f4 buf[2048];                       // 32 KB staging (of 320 KB WGP LDS)
  const int tid   = threadIdx.x;
  const int chunk = blockIdx.x;
  const int plane = chunk >> 3;                  // 8 chunks per plane
  const int rq0   = (chunk & 7) * 2048;          // first quad within plane
  const float* __restrict__ src = in  + plane * IN_PLANE;
  float*       __restrict__ dst = out + plane * OUT_PLANE;

  // Stage 1: 8 async B128 loads per lane, global -> LDS (tracked by ASYNCcnt).
  // Input side is 16B aligned (row stride 1024B, plane stride 256KB).
  // Streaming data (touched once, ~200MB vs 192MB L2) -> non-temporal hint.
#pragma unroll
  for (int k = 0; k < 8; ++k) {
    const int rq = rq0 + k * 256 + tid;          // quad index within plane
    const float* gsrc = src + rq * 4;            // 16B aligned
    unsigned lds = (unsigned)(unsigned long long)(&buf[k * 256 + tid]);
    asm volatile("global_load_async_to_lds_b128 %0, %1, off th:TH_LOAD_NT"
                 :: "v"(lds), "v"(gsrc)
                 : "memory");
  }

  // Stage 2: pipelined drain. All ASYNCcnt ops in this wave are loads and
  // async loads complete in order, so asynccnt <= (7-k) guarantees the k-th
  // load has landed in LDS. Each lane reads only slots its own wave filled,
  // so no workgroup barrier is needed. Output is shifted by +1 col -> only
  // 4B aligned; memcpy keeps codegen legal (b128 store, 4B align).
#define DRAIN_STEP(k, rem)                                            \
  {                                                                   \
    asm volatile("s_wait_asynccnt " #rem ::: "memory");               \
    f4 v = buf[(k) * 256 + tid];                /* ds_load_b128 */    \
    const int rq = rq0 + (k) * 256 + tid;                             \
    float* o = dst + ((rq >> 6) + 1) * WP + 1 + (rq & 63) * 4;        \
    __builtin_memcpy(o, &v, 16);                /* global_store_b128 */\
  }
  DRAIN_STEP(0, 7)
  DRAIN_STEP(1, 6)
  DRAIN_STEP(2, 5)
  DRAIN_STEP(3, 4)
  DRAIN_STEP(4, 3)
  DRAIN_STEP(5, 2)
  DRAIN_STEP(6, 1)
  DRAIN_STEP(7, 0)
#undef DRAIN_STEP
}

// ---------------------------------------------------------------------------
// Halo ring: 1028 elements per plane, gathered from neighbor faces.
// Face order: 0=front 1=right 2=back 3=left 4=top 5=down.
// Corners replicate the adjacent top/down-plate value (edge() with p=1),
// implemented as a clamp on the top/bottom-row column index.
// ---------------------------------------------------------------------------
__global__ __launch_bounds__(256) void cubepad_halo(
    const float* __restrict__ in, float* __restrict__ out) {
  const int idx = blockIdx.x * 256 + threadIdx.x;
  if (idx >= NPLANES * RING) return;
  const int p  = idx / RING;
  const int e  = idx - p * RING;
  const int c  = p & 63;          // channel
  const int nf = p >> 6;
  const int n  = nf / 6;          // cube (batch) index
  const int f  = nf - n * 6;      // face index

  int g = 0, r = 0, col = 0;      // source face, row, col
  int oidx;

  if (e < 516) {                  // top (e<258) / bottom (258<=e<516) rows, full width
    const bool top = (e < 258);
    const int  x   = top ? e : e - 258;
    int w = x - 1; w = w < 0 ? 0 : (w > 255 ? 255 : w);   // corner clamp
    if (top) {
      switch (f) {
        case 0: g = 4; r = 255;     col = w;        break;  // front <- top bottom row
        case 1: g = 4; r = 255 - w; col = 255;      break;
        case 2: g = 4; r = 0;       col = 255 - w;  break;
        case 3: g = 4; r = w;       col = 0;        break;
        case 4: g = 2; r = 0;       col = 255 - w;  break;  // top <- back (flipped)
        default:g = 0; r = 255;     col = w;        break;  // down <- front
      }
      oidx = p * OUT_PLANE + x;                    // y = 0
    } else {
      switch (f) {
        case 0: g = 5; r = 0;       col = w;        break;
        case 1: g = 5; r = w;       col = 255;      break;
        case 2: g = 5; r = 255;     col = 255 - w;  break;
        case 3: g = 5; r = 255 - w; col = 0;        break;
        case 4: g = 0; r = 0;       col = w;        break;
        default:g = 2; r = 255;     col = 255 - w;  break;
      }
      oidx = p * OUT_PLANE + 257 * WP + x;         // y = 257
    }
  } else {                        // left (e<772) / right columns, y in [1,256]
    const bool left = (e < 772);
    const int  h    = left ? e - 516 : e - 772;
    if (left) {
      switch (f) {
        case 0: g = 3; r = h;   col = 255;      break;
        case 1: g = 0; r = h;   col = 255;      break;
        case 2: g = 1; r = h;   col = 255;      break;
        case 3: g = 2; r = h;   col = 255;      break;
        case 4: g = 3; r = 0;   col = h;        break;
        default:g = 3; r = 255; col = 255 - h;  break;
      }
      oidx = p * OUT_PLANE + (h + 1) * WP;         // x = 0
    } else {
      switch (f) {
        case 0: g = 1; r = h;   col = 0;        break;
        case 1: g = 2; r = h;   col = 0;        break;
        case 2: g = 3; r = h;   col = 0;        break;
        case 3: g = 0; r = h;   col = 0;        break;
        case 4: g = 1; r = 0;   col = 255 - h;  break;
        default:g = 1; r = 255; col = h;        break;
      }
      oidx = p * OUT_PLANE + (h + 1) * WP + 257;   // x = 257
    }
  }

  const int src_plane = (n * 6 + g) * C + c;
  out[oidx] = in[src_plane * IN_PLANE + r * W + col];
}

extern "C" void kernel_launch(void* const* d_in, const int* in_sizes, int n_in,
                              void* d_out, int out_size, void* d_ws, size_t ws_size,
                              hipStream_t stream) {
  (void)in_sizes; (void)n_in; (void)d_ws; (void)ws_size; (void)out_size;
  const float* x   = (const float*)d_in[0];
  float*       out = (float*)d_out;

  // Interior: 768 planes * 8 chunks = 6144 blocks, 256 threads each.
  cubepad_interior<<<6144, 256, 0, stream>>>(x, out);
  // Halo ring: 768 * 1028 = 789504 elements -> 3084 blocks.
  cubepad_halo<<<(NPLANES * RING + 255) / 256, 256, 0, stream>>>(x, out);
}